// TopKOperator_54331336294469
// MI455X (gfx1250) — compile-verified
//
#include <hip/hip_runtime.h>
#include <stdint.h>

#define BATCH 8
#define L0    8192
#define EMB   768
#define DEPTH 4
#define OUTL  512
#define NTHR  1024

typedef float v2f __attribute__((ext_vector_type(2)));
typedef float v8f __attribute__((ext_vector_type(8)));
typedef unsigned int v4u __attribute__((ext_vector_type(4)));
typedef int v4i __attribute__((ext_vector_type(4)));
typedef int v8i __attribute__((ext_vector_type(8)));

// TDM builtin arity differs across toolchains: ROCm 7.2 clang-22 = 5 args; the
// amdgpu-toolchain (which uniquely ships the TDM header) = 6 args (extra zero v8i ok).
#if __has_include(<hip/amd_detail/amd_gfx1250_TDM.h>)
#define TDM_GATHER(g0, g1, g2, g3) \
  __builtin_amdgcn_tensor_load_to_lds((g0), (g1), (g2), (g3), (v8i){0,0,0,0,0,0,0,0}, 0)
#else
#define TDM_GATHER(g0, g1, g2, g3) \
  __builtin_amdgcn_tensor_load_to_lds((g0), (g1), (g2), (g3), 0)
#endif

// Order-preserving float32 <-> u32 transform, descending (higher score -> smaller key).
__device__ __forceinline__ unsigned enc_desc(float f) {
  unsigned u = __float_as_uint(f);
  u ^= (u >> 31) ? 0xFFFFFFFFu : 0x80000000u;   // ascending transform
  return ~u;                                     // flip -> descending
}
__device__ __forceinline__ float dec_desc(unsigned e) {
  unsigned up = ~e;
  unsigned u  = (up >> 31) ? (up ^ 0x80000000u) : ~up;
  return __uint_as_float(u);
}

// Block-wide exclusive scan: wave32 shuffle scans + 32-entry wave-sum scan (2 barriers).
__device__ __forceinline__ int block_excl_scan(int s, int tid, int* s_wsum) {
  __syncthreads();                     // protect s_wsum reuse across calls
  const int lane = tid & 31, wid = tid >> 5;
  int inc = s;
#pragma unroll
  for (int d = 1; d < 32; d <<= 1) {
    int n = __shfl_up(inc, d, 32);
    if (lane >= d) inc += n;
  }
  if (lane == 31) s_wsum[wid] = inc;
  __syncthreads();
  if (wid == 0) {
    int w = s_wsum[lane];
#pragma unroll
    for (int d = 1; d < 32; d <<= 1) {
      int n = __shfl_up(w, d, 32);
      if (lane >= d) w += n;
    }
    s_wsum[lane] = w;
  }
  __syncthreads();
  int base = (wid > 0) ? s_wsum[wid - 1] : 0;
  return base + inc - s;
}

// One stable LSD radix pass (4-bit digit at `shift`) of N u64 elements, src -> dst.
// Ownership is round-major strided (elem j = e*NTHR + tid) so (round, wave, lane)
// lexicographic order == array order; in-wave ranks come from ballot match-masks and
// per-(digit, round*32+wave) counts live in an 8 KB matrix scanned digit-major.
__device__ __forceinline__ void radix_pass(const unsigned long long* src,
                                           unsigned long long* dst, int N, int shift,
                                           unsigned short* cnt, int* s_wsum, int tid) {
  const int rounds = N >> 10;           // N / NTHR  (8,4,2,1)
  const int lane = tid & 31, wid = tid >> 5;
  const int ncols = rounds * 32;
  for (int f = tid; f < ncols * 16; f += NTHR) cnt[f] = 0;
  __syncthreads();

  unsigned long long vals[8];
  int dr[8];                            // digit | (in-wave rank << 8)
  for (int e = 0; e < rounds; ++e) {
    unsigned long long v = src[e * NTHR + tid];
    int d = (int)((v >> shift) & 0xF);
    unsigned long long m0 = __ballot(d & 1);
    unsigned long long m1 = __ballot(d & 2);
    unsigned long long m2 = __ballot(d & 4);
    unsigned long long m3 = __ballot(d & 8);
    unsigned mm = (unsigned)(((d & 1) ? m0 : ~m0) & ((d & 2) ? m1 : ~m1) &
                             ((d & 4) ? m2 : ~m2) & ((d & 8) ? m3 : ~m3));
    int rank = __popc(mm & ((1u << lane) - 1u));
    if (rank == 0) cnt[d * ncols + (e * 32 + wid)] = (unsigned short)__popc(mm);
    vals[e] = v;
    dr[e] = d | (rank << 8);
  }
  __syncthreads();

  {   // exclusive scan over cnt[0 .. ncols*16)
    int M = ncols * 16;                                   // 4096..512
    int chunk = (M + NTHR - 1) / NTHR;
    int base = tid * chunk;
    int s = 0;
    for (int e = 0; e < chunk; ++e) { int f = base + e; if (f < M) s += cnt[f]; }
    int run = block_excl_scan(s, tid, s_wsum);
    for (int e = 0; e < chunk; ++e) {
      int f = base + e;
      if (f < M) { int t = cnt[f]; cnt[f] = (unsigned short)run; run += t; }
    }
  }
  __syncthreads();

  for (int e = 0; e < rounds; ++e) {
    int d = dr[e] & 0xFF, rank = dr[e] >> 8;
    dst[cnt[d * ncols + (e * 32 + wid)] + rank] = vals[e];
  }
  __syncthreads();
}

// Even number of passes -> result lands back in bufA.
__device__ __forceinline__ void radix_sort(unsigned long long* bufA, unsigned long long* bufB,
                                           int N, int npasses, unsigned short* cnt,
                                           int* s_wsum, int tid) {
  for (int p = 0; p < npasses; p += 2) {
    radix_pass(bufA, bufB, N, 32 + 4 * p, cnt, s_wsum, tid);
    radix_pass(bufB, bufA, N, 32 + 4 * (p + 1), cnt, s_wsum, tid);
  }
}

// ---------------- Phase 1: score chain + composition table (1 workgroup / batch) --------------
__global__ __launch_bounds__(NTHR) void topk_scores_kernel(
    const float* __restrict__ scores, float* __restrict__ out_scores,
    int* __restrict__ comp_idx, float* __restrict__ comp_w) {
  __shared__ unsigned long long bufA[L0];        // 64 KB  keys: (desc-score | row)
  __shared__ union {
    unsigned long long bufB[L0];                 // 64 KB  radix ping-pong buffer
    struct {                                     // merge-phase overlays (sorts are done then)
      unsigned short pfx[L0];                    // 16 KB  SORT_BACK rank prefix
      unsigned short newrow[L0];                 // 16 KB  per-old-row new row id
      float          mult[L0];                   // 32 KB  per-old-row weight multiplier
    } mg;
  } uB;
  __shared__ unsigned short s_cnt[4096];         //  8 KB  radix (digit x column) counts
  __shared__ int s_wsum[32];

  const int b   = blockIdx.x;
  const int tid = threadIdx.x;

  // Each thread privately tracks 8 original indices: current row + accumulated weight.
  int   currow[L0 / NTHR];
  float accw[L0 / NTHR];
  for (int q = 0; q < L0 / NTHR; ++q) {
    int o = tid + q * NTHR;
    currow[q] = o;
    accw[q]   = 1.0f;
    bufA[o] = ((unsigned long long)enc_desc(scores[(size_t)b * L0 + o]) << 32) | (unsigned)o;
  }
  __syncthreads();

  int L = L0;
  for (int lev = 0; lev < DEPTH; ++lev) {
    // Stable ascending radix on the 32-bit key == argsort(-score) with index tie-break.
    radix_sort(bufA, uB.bufB, L, 8, s_cnt, s_wsum, tid);
    const int half = L >> 1;

    // Mark "left" rows (top half) then exclusive-scan over the row-id domain:
    // pos = rank of a left row-id among left row-ids == SORT_BACK destination.
    for (int v = tid; v < L; v += NTHR) uB.mg.pfx[v] = 0;
    __syncthreads();
    for (int i = tid; i < half; i += NTHR)
      uB.mg.pfx[(unsigned)(bufA[i] & 0xFFFFFFFFu)] = 1;
    __syncthreads();
    {
      int chunk = L / NTHR;                      // L >= 1024 -> >= 1, exact
      int base = tid * chunk;
      int s = 0;
      for (int e = 0; e < chunk; ++e) s += uB.mg.pfx[base + e];
      int run = block_excl_scan(s, tid, s_wsum);
      for (int e = 0; e < chunk; ++e) {
        int t = uB.mg.pfx[base + e];
        uB.mg.pfx[base + e] = (unsigned short)run;
        run += t;
      }
      __syncthreads();
    }

    // Merge phase A: pair rank i with rank L-1-i (FLIP_RIGHT), softmax(2^sc) weights.
    float ns_s[4]; int pos_s[4]; int cnt = 0;
    for (int i = tid; i < half; i += NTHR) {
      unsigned long long kl = bufA[i], kr = bufA[L - 1 - i];
      int l = (int)(kl & 0xFFFFFFFFu), r = (int)(kr & 0xFFFFFFFFu);
      float sc0 = dec_desc((unsigned)(kl >> 32));
      float sc1 = dec_desc((unsigned)(kr >> 32));
      float p0 = exp2f(sc0), p1 = exp2f(sc1);
      float m  = fmaxf(p0, p1);
      float e0 = expf(p0 - m), e1 = expf(p1 - m);
      float inv = 1.0f / (e0 + e1);
      float w0 = e0 * inv, w1 = e1 * inv;
      float ns = sc0 * w0 + sc1 * w1;
      int pos = uB.mg.pfx[l];
      uB.mg.mult[l] = w0; uB.mg.newrow[l] = (unsigned short)pos;
      uB.mg.mult[r] = w1; uB.mg.newrow[r] = (unsigned short)pos;
      ns_s[cnt] = ns; pos_s[cnt] = pos; ++cnt;
    }
    __syncthreads();
    // Phase B: next-level keys (row id == position); remap privately tracked originals.
    for (int c = 0; c < cnt; ++c)
      bufA[pos_s[c]] = ((unsigned long long)enc_desc(ns_s[c]) << 32) | (unsigned)pos_s[c];
    for (int q = 0; q < L0 / NTHR; ++q) {
      int rr = currow[q];
      accw[q] *= uB.mg.mult[rr];
      currow[q] = uB.mg.newrow[rr];
    }
    __syncthreads();
    L = half;
  }

  // Final scores: bufA[p] holds (score_p | p) for p in [0,512).
  if (tid < OUTL)
    out_scores[b * OUTL + tid] = dec_desc((unsigned)(bufA[tid] >> 32));
  __syncthreads();

  // Deterministic composition table: radix-sort (final_row<<13 | orig_idx) keys (22 bits,
  // 6 passes). Each final row owns exactly 16 originals, so sorted position g maps to
  // (row = g>>4, slot = g&15) with ascending-index slot order — no atomics anywhere.
  // Ascending slot order also satisfies the TDM gather-mode index-monotonicity rule.
  for (int q = 0; q < L0 / NTHR; ++q) {
    int o = tid + q * NTHR;
    unsigned key = ((unsigned)currow[q] << 13) | (unsigned)o;
    bufA[o] = ((unsigned long long)key << 32) | __float_as_uint(accw[q]);
  }
  __syncthreads();
  radix_sort(bufA, uB.bufB, L0, 6, s_cnt, s_wsum, tid);
  for (int g = tid; g < L0; g += NTHR) {
    unsigned long long kk = bufA[g];
    int o   = (int)((kk >> 32) & 0x1FFFu);
    float w = __uint_as_float((unsigned)(kk & 0xFFFFFFFFu));
    size_t idx = ((size_t)b * OUTL + (g >> 4)) * 16 + (g & 15);
    comp_idx[idx] = o;
    comp_w[idx]   = w;
  }
}

// ---------------- Phase 2: TDM row-gather into LDS + V_WMMA_F32_16X16X4_F32 combine ----------
// out[j][c] = sum_{k<16} w_k * emb[idx_k][c].  One block (4 waves) per output row.
// Wave 0 issues a single TENSOR_LOAD_TO_LDS in gather mode: 16 x 16-bit row indices into
// the [8192 x 768] f32 tensor, tile width 768, DMA'd as contiguous 3 KB streams into LDS.
// Waves then run the WMMA combine with B operands from LDS: A = weight vector broadcast
// over all 16 M rows, B = gathered emb columns, C accumulates f32.
__global__ __launch_bounds__(128) void topk_embs_kernel(
    const float* __restrict__ embs, const int* __restrict__ comp_idx,
    const float* __restrict__ comp_w, float* __restrict__ out_embs) {
  __shared__ float g_rows[16 * EMB];   // 48 KB gathered contribution rows
  __shared__ float l_w[16];
  const int blk = blockIdx.x;
  const int b   = blk >> 9;     // /512
  const int j   = blk & 511;
  const int tid = threadIdx.x;
  const size_t ci = ((size_t)b * OUTL + j) * 16;
  if (tid < 16) l_w[tid] = comp_w[ci + tid];

  if (tid < 32) {               // wave 0 builds the D# and issues the TDM gather
    unsigned long long gaddr =
        (unsigned long long)reinterpret_cast<uintptr_t>(embs + (size_t)b * L0 * EMB);
    unsigned lds_off = (unsigned)reinterpret_cast<uintptr_t>(&g_rows[0]); // low32 = LDS byte addr
    v4u g0;
    g0[0] = 0x80000001u;                                   // count=1 | gather_mode (16-bit idx)
    g0[1] = lds_off;                                       // lds_addr
    g0[2] = (unsigned)gaddr;                               // global_addr[31:0]
    g0[3] = ((unsigned)(gaddr >> 32) & 0x01FFFFFFu) | (2u << 30);  // addr[56:32] | type=2
    v8i g1;
    g1[0] = (int)(2u << 16);        // workgroup_mask=0 | data_size=4B
    g1[1] = (int)((unsigned)EMB << 16);   // atomic_barrier=0 | tensor_dim0[15:0]=768
    g1[2] = (int)((unsigned)L0 << 16);    // tensor_dim0[31:16]=0 | tensor_dim1[15:0]=8192
    g1[3] = (int)((unsigned)EMB << 16);   // tensor_dim1[31:16]=0 | tile_dim0=768
    g1[4] = 16;                     // tile_dim1 = #valid gather indices
    g1[5] = EMB;                    // tensor_dim0_stride[31:0] = 768 elements
    g1[6] = 0;                      // stride hi | dim1_stride (ignored in gather mode)
    g1[7] = 0;
    v4i g2, g3;                     // 16 packed 16-bit row indices (non-decreasing)
#pragma unroll
    for (int p = 0; p < 4; ++p) {
      int a0 = comp_idx[ci + 2 * p],     a1 = comp_idx[ci + 2 * p + 1];
      int b0 = comp_idx[ci + 8 + 2 * p], b1 = comp_idx[ci + 9 + 2 * p];
      g2[p] = __builtin_amdgcn_readfirstlane(a0 | (a1 << 16));
      g3[p] = __builtin_amdgcn_readfirstlane(b0 | (b1 << 16));
    }
    TDM_GATHER(g0, g1, g2, g3);
    __builtin_amdgcn_s_wait_tensorcnt(0);
  }
  __syncthreads();                  // publish gathered rows + weights to all 4 waves

  const int wv   = tid >> 5;
  const int lane = tid & 31;
  const int hi   = lane >> 4;       // half-wave selects K={0,1} vs K={2,3} slots
  const int n    = lane & 15;       // column within tile
  float* out_row = out_embs + ((size_t)b * OUTL + j) * EMB;

  for (int tt = 0; tt < 12; ++tt) {
    int cb = (wv * 12 + tt) * 16;
    v8f c = {0.f, 0.f, 0.f, 0.f, 0.f, 0.f, 0.f, 0.f};
#pragma unroll
    for (int k0 = 0; k0 < 16; k0 += 4) {
      int kx = k0 + (hi << 1);
      v2f a;  a.x  = l_w[kx];                       // A[m][k] = w[k] (broadcast over m)
              a.y  = l_w[kx + 1];
      v2f bv; bv.x = g_rows[kx * EMB + cb + n];     // B[k][n] = emb[idx_k][cb+n] from LDS
              bv.y = g_rows[(kx + 1) * EMB + cb + n];
      c = __builtin_amdgcn_wmma_f32_16x16x4_f32(false, a, false, bv, (short)0, c, false, false);
    }
    if (lane < 16) out_row[cb + n] = c[0];          // all M rows equal; take M=0
  }
}

extern "C" void kernel_launch(void* const* d_in, const int* in_sizes, int n_in,
                              void* d_out, int out_size, void* d_ws, size_t ws_size,
                              hipStream_t stream) {
  (void)in_sizes; (void)n_in; (void)out_size; (void)ws_size;
  const float* embs   = (const float*)d_in[0];
  const float* scores = (const float*)d_in[1];
  float* out_embs   = (float*)d_out;
  float* out_scores = out_embs + (size_t)BATCH * OUTL * EMB;
  int*   comp_idx = (int*)d_ws;
  float* comp_w   = (float*)((char*)d_ws + (size_t)BATCH * OUTL * 16 * sizeof(int));

  topk_scores_kernel<<<dim3(BATCH), dim3(NTHR), 0, stream>>>(scores, out_scores, comp_idx, comp_w);
  topk_embs_kernel<<<dim3(BATCH * OUTL), dim3(128), 0, stream>>>(embs, comp_idx, comp_w, out_embs);
}